// HAFS_Loss_64871186039053
// MI455X (gfx1250) — compile-verified
//
#include <hip/hip_runtime.h>
#include <math.h>

#define DIMV 435
#define LEAFN 250
#define BATCH 64
#define NPAIRS 14            // 28 N-tiles of 16 cols -> 14 pairs of 32 cols

typedef __attribute__((ext_vector_type(2))) float v2f;
typedef __attribute__((ext_vector_type(8))) float v8f;

#define WMMA_F32(A, B, C) \
    __builtin_amdgcn_wmma_f32_16x16x4_f32(false, (A), false, (B), (short)0, (C), false, false)

// Computes a full-M (64) x 32-column pair of tiles of  Y = X (64x435) @ Bm^T,
// Bm row-major with row index = output column c, col index = d.
// accA = columns [c0, c0+16), accB = columns [c0+16, c0+32).
// A-fragments (from cache-resident X) are shared by both tiles: 8 WMMAs per
// K-step for 8 A-loads + 4 B-loads.
__device__ __forceinline__ void gemm_pair_f32(
    const float* __restrict__ X,
    const float* __restrict__ Bm,
    int c0, int lane, v8f accA[4], v8f accB[4])
{
    const int ml  = lane & 15;          // M (A) / N (B) index within tile
    const int klo = (lane >> 4) << 1;   // lanes 0-15 hold K={0,1}, lanes 16-31 K={2,3}
    const int ca  = c0 + ml;
    const int cb  = c0 + 16 + ml;
    const bool va = (ca < DIMV);
    const bool vb = (cb < DIMV);
    const float* __restrict__ bpa = Bm + (size_t)(va ? ca : (DIMV - 1)) * DIMV;
    const float* __restrict__ bpb = Bm + (size_t)(vb ? cb : (DIMV - 1)) * DIMV;

    // main K loop: d = 0..431 in steps of 4 (all in-bounds)
    #pragma unroll 4
    for (int kb = 0; kb < 432; kb += 4) {
        const int kc = kb + klo;
        v2f bfa, bfb;
        bfa.x = bpa[kc];
        bfa.y = bpa[kc + 1];
        bfb.x = bpb[kc];
        bfb.y = bpb[kc + 1];
        if (!va) { bfa.x = 0.f; bfa.y = 0.f; }
        if (!vb) { bfb.x = 0.f; bfb.y = 0.f; }
        #pragma unroll
        for (int t = 0; t < 4; ++t) {
            const float* __restrict__ ap = X + (size_t)(t * 16 + ml) * DIMV + kc;
            v2f af;
            af.x = ap[0];
            af.y = ap[1];
            accA[t] = WMMA_F32(af, bfa, accA[t]);
            accB[t] = WMMA_F32(af, bfb, accB[t]);
        }
    }

    // tail: kb = 432 covers d = 432..434 (d = 435 zero-padded)
    {
        const int kc = 432 + klo;        // 432 (lanes 0-15) or 434 (lanes 16-31)
        const bool k1 = (kc + 1 < DIMV);
        v2f bfa, bfb;
        bfa.x = bpa[kc];
        bfa.y = k1 ? bpa[kc + 1] : 0.f;
        bfb.x = bpb[kc];
        bfb.y = k1 ? bpb[kc + 1] : 0.f;
        if (!va) { bfa.x = 0.f; bfa.y = 0.f; }
        if (!vb) { bfb.x = 0.f; bfb.y = 0.f; }
        #pragma unroll
        for (int t = 0; t < 4; ++t) {
            const float* __restrict__ ap = X + (size_t)(t * 16 + ml) * DIMV + kc;
            v2f af;
            af.x = ap[0];
            af.y = k1 ? ap[1] : 0.f;
            accA[t] = WMMA_F32(af, bfa, accA[t]);
            accB[t] = WMMA_F32(af, bfb, accB[t]);
        }
    }
}

// Kernel 1: out[b,n] = || X @ P_n^T ||_row  -> d_out[1 + b*250 + n]
// One block per leaf n; streams P_n (189 MB total across the grid) exactly once.
__global__ __launch_bounds__(256) void hafs_dist_kernel(
    const float* __restrict__ logits,
    const float* __restrict__ proj,
    float* __restrict__ out)
{
    __shared__ float rowsq[BATCH];
    const int n    = blockIdx.x;
    const int tid  = threadIdx.x;
    const int wave = tid >> 5;
    const int lane = tid & 31;

    if (tid < BATCH) rowsq[tid] = 0.f;
    __syncthreads();

    const float* __restrict__ Bm = proj + (size_t)n * DIMV * DIMV;

    v8f sq[4] = {};
    for (int p = wave; p < NPAIRS; p += 8) {
        v8f accA[4] = {};
        v8f accB[4] = {};
        gemm_pair_f32(logits, Bm, p * 32, lane, accA, accB);
        #pragma unroll
        for (int t = 0; t < 4; ++t) {
            sq[t] += accA[t] * accA[t];
            sq[t] += accB[t] * accB[t];
        }
    }

    // D-tile layout: VGPR r -> (M=r, N=lane) lanes 0-15, (M=r+8, N=lane-16) lanes 16-31.
    // Sum squares across the 16 lanes of each half, combine waves through LDS.
    #pragma unroll
    for (int t = 0; t < 4; ++t) {
        #pragma unroll
        for (int r = 0; r < 8; ++r) {
            float v = sq[t][r];
            v += __shfl_xor(v, 1);
            v += __shfl_xor(v, 2);
            v += __shfl_xor(v, 4);
            v += __shfl_xor(v, 8);
            if (lane == 0)  atomicAdd(&rowsq[t * 16 + r], v);
            if (lane == 16) atomicAdd(&rowsq[t * 16 + r + 8], v);
        }
    }
    __syncthreads();

    if (tid < BATCH) out[1 + (size_t)tid * LEAFN + n] = sqrtf(rowsq[tid]);
}

// Kernel 2: ws[b*435 + c] = | (X @ basis^T)[b,c] |   (same WMMA core, one block)
__global__ __launch_bounds__(256) void hafs_absproj_kernel(
    const float* __restrict__ logits,
    const float* __restrict__ basis,
    float* __restrict__ ws)
{
    const int tid  = threadIdx.x;
    const int wave = tid >> 5;
    const int lane = tid & 31;
    const int ml   = lane & 15;
    const int mh   = (lane >> 4) << 3;

    for (int p = wave; p < NPAIRS; p += 8) {
        v8f accA[4] = {};
        v8f accB[4] = {};
        gemm_pair_f32(logits, basis, p * 32, lane, accA, accB);

        const int colA = p * 32 + ml;
        const int colB = p * 32 + 16 + ml;
        #pragma unroll
        for (int t = 0; t < 4; ++t) {
            #pragma unroll
            for (int r = 0; r < 8; ++r) {
                const size_t row = (size_t)(t * 16 + mh + r);
                if (colA < DIMV) ws[row * DIMV + colA] = fabsf(accA[t][r]);
                if (colB < DIMV) ws[row * DIMV + colB] = fabsf(accB[t][r]);
            }
        }
    }
}

// Kernel 3: scalar loss -> d_out[0]. One thread per batch row (64 threads).
__global__ __launch_bounds__(64) void hafs_loss_kernel(
    const float* __restrict__ logits,
    const float* __restrict__ node_prob,
    const int* __restrict__ labels,
    const int* __restrict__ hier,
    const float* __restrict__ absproj,
    float* __restrict__ out)
{
    __shared__ float s_kl;
    __shared__ float s_pos[4], s_npos[4], s_neg[4], s_nneg[4];
    __shared__ int   s_any[4];

    const int b = threadIdx.x;
    if (b < 4) { s_pos[b] = 0.f; s_npos[b] = 0.f; s_neg[b] = 0.f; s_nneg[b] = 0.f; s_any[b] = 0; }
    if (b == 0) s_kl = 0.f;
    __syncthreads();

    const int offs[4] = {0, 10, 60, 185};
    const int nls[4]  = {10, 50, 125, 250};

    const int lab = labels[b];
    const float* __restrict__ xb = logits + (size_t)b * DIMV;

    // log-sum-exp of |x_b|
    float mx = 0.f;
    for (int d = 0; d < DIMV; ++d) mx = fmaxf(mx, fabsf(xb[d]));
    float se = 0.f;
    for (int d = 0; d < DIMV; ++d) se += expf(fabsf(xb[d]) - mx);
    const float lse = mx + logf(se);

    float kl_row = 0.f;
    #pragma unroll
    for (int l = 0; l < 4; ++l) {
        const int off = offs[l];
        const int nl  = nls[l];
        const int ll  = hier[lab * 4 + l];
        const bool pos = (ll >= 0);
        const float* __restrict__ apb = absproj + (size_t)b * DIMV + off;

        float nrm2 = 0.f, sabs = 0.f;
        for (int j = 0; j < nl; ++j) { const float v = apb[j]; nrm2 += v * v; sabs += v; }
        const float inv    = 1.0f / sqrtf(nrm2);
        const int   lc     = pos ? ll : 0;                     // clip(label, 0)
        const float coslab = apb[lc] * inv;
        // sum_j |ohe_j - cos_j| with cos_j >= 0
        const float rowterm = (sabs * inv - coslab) + fabsf(1.f - coslab);

        if (pos) { atomicAdd(&s_pos[l], rowterm); atomicAdd(&s_npos[l], 1.f); }
        else     { atomicAdd(&s_neg[l], sabs);    atomicAdd(&s_nneg[l], 1.f); }
        if (ll > 0) atomicOr(&s_any[l], 1);

        // KL: l_gt row has a single nonzero per level at column off+lc with mass g
        const float g     = pos ? node_prob[lab * 4 + l] : 0.f;
        const float xl    = fabsf(xb[off + lc]);
        const float xlogy = (g > 0.f) ? g * logf(g) : 0.f;
        kl_row += xlogy - g * (xl - lse);
    }
    atomicAdd(&s_kl, kl_row);
    __syncthreads();

    if (b == 0) {
        float laux = 0.f;
        #pragma unroll
        for (int l = 0; l < 4; ++l) {
            if (s_any[l])        laux += s_pos[l] / fmaxf(s_npos[l], 1.f);
            if (s_nneg[l] > 0.f) laux += s_neg[l] / fmaxf(s_nneg[l], 1.f);
        }
        out[0] = s_kl * (1.0f / 64.0f) + 0.1f * laux;
    }
}

extern "C" void kernel_launch(void* const* d_in, const int* in_sizes, int n_in,
                              void* d_out, int out_size, void* d_ws, size_t ws_size,
                              hipStream_t stream) {
    const float* logits = (const float*)d_in[0];   // [64,435]
    const float* proj   = (const float*)d_in[1];   // [250,435,435]
    const float* basis  = (const float*)d_in[2];   // [435,435]
    const float* nprob  = (const float*)d_in[3];   // [250,4]
    const int*   labels = (const int*)d_in[4];     // [64]
    const int*   hier   = (const int*)d_in[5];     // [250,4]
    float* out = (float*)d_out;                    // [1 + 64*250]
    float* ws  = (float*)d_ws;                     // needs 64*435 floats

    hipLaunchKernelGGL(hafs_absproj_kernel, dim3(1), dim3(256), 0, stream,
                       logits, basis, ws);
    hipLaunchKernelGGL(hafs_dist_kernel, dim3(LEAFN), dim3(256), 0, stream,
                       logits, proj, out);
    hipLaunchKernelGGL(hafs_loss_kernel, dim3(1), dim3(64), 0, stream,
                       logits, nprob, labels, hier, ws, out);
}